// EnhancedGAT_20693152432872
// MI455X (gfx1250) — compile-verified
//
#include <hip/hip_runtime.h>

// =====================================================================
// EnhancedGAT pipeline for MI455X (gfx1250, wave32).
// GEMMs: W staged in LDS as k-pairs (ds_load_b64 B fragments), per-lane
// A strip prefetched to registers, fully-unrolled fp32 WMMA
// (V_WMMA_F32_16X16X4_F32) with CT accumulators per wave.
// Graph scatters: wave-per-edge coalesced float atomics (HBM-bound part).
// =====================================================================

typedef float v2f __attribute__((ext_vector_type(2)));
typedef float v8f __attribute__((ext_vector_type(8)));

static constexpr int Nn = 50000;
static constexpr int Ee = 800000;
static constexpr int Ff = 128;
static constexpr int Hh = 96;
static constexpr int Cc = 40;

#define NEG_SLOPE 0.2f
#define LN_EPS 1e-5f

__device__ __forceinline__ float eluf(float x) {
  return x > 0.f ? x : (__expf(x) - 1.f);
}

// Monotonic float atomic-max via int (>=0) / uint (<0) ordering trick.
__device__ __forceinline__ void atomicMaxF(float* addr, float val) {
  if (val >= 0.f) atomicMax((int*)addr, __float_as_int(val));
  else            atomicMin((unsigned int*)addr, (unsigned int)__float_as_int(val));
}

// ---------------------------------------------------------------------
// GEMM: Y[nrows,M] = X[nrows,K] @ W[K,M] (+bias1 +bias2) (+=Y if accumulate)
// Template: CT = M tiles (M padded to CT*16), K = reduction dim (mult of 4).
// Block = 256 threads = 8 waves; W staged once into LDS as float2 k-pairs:
//   wlds[kp*Mpad + c] = { W[2kp][c], W[2kp+1][c] }
// so lane l's B fragment (k = 2*(l>>4)+v, n = l&15) is one ds_load_b64.
// A frag (16x4): lane m=l&15, 2 VGPRs, k = 2*(l>>4)+v  -> one b64/K-step,
// entire strip prefetched into registers before the WMMA chain.
// C/D (16x16): vgpr r, lane l -> m = 8*(l>>4)+r, n = l&15.
// OOB rows/cols clamped for loads; stores guarded in epilogue.
// ---------------------------------------------------------------------
template <int CT, int K>
__global__ __launch_bounds__(256) void gemm_wmma_f32(
    const float* __restrict__ X, const float* __restrict__ W,
    float* __restrict__ Y,
    const float* __restrict__ bias1, const float* __restrict__ bias2,
    int nrows, int M, int accumulate) {
  constexpr int Mpad = CT * 16;
  constexpr int K4   = K / 4;
  __shared__ float2 wlds[(K / 2) * Mpad];

  // Cooperative stage of W into LDS (k-pair layout, clamped columns).
  for (int idx = threadIdx.x; idx < (K / 2) * Mpad; idx += blockDim.x) {
    const int kp = idx / Mpad;
    const int c  = idx - kp * Mpad;
    const int cc = (c < M) ? c : (M - 1);
    float2 wv;
    wv.x = W[(size_t)(2 * kp) * M + cc];
    wv.y = W[(size_t)(2 * kp + 1) * M + cc];
    wlds[idx] = wv;
  }
  __syncthreads();

  const int lane = threadIdx.x & 31;
  const int tm   = (int)((blockIdx.x * (unsigned)blockDim.x + threadIdx.x) >> 5);
  const int rowTiles = (nrows + 15) >> 4;
  if (tm >= rowTiles) return;

  const int half = lane >> 4;   // 0 or 1
  const int l16  = lane & 15;
  int arow = tm * 16 + l16;
  if (arow >= nrows) arow = nrows - 1;          // clamp: loads always valid

  int bcol[CT];
#pragma unroll
  for (int t = 0; t < CT; ++t) bcol[t] = t * 16 + l16;   // valid vs Mpad

  v8f acc[CT];
  if (accumulate) {
#pragma unroll
    for (int t = 0; t < CT; ++t) {
      const int cc = (bcol[t] < M) ? bcol[t] : (M - 1);
#pragma unroll
      for (int r = 0; r < 8; ++r) {
        int row = tm * 16 + half * 8 + r;
        if (row >= nrows) row = nrows - 1;      // clamp (unstored anyway)
        acc[t][r] = Y[(size_t)row * M + cc];
      }
    }
  } else {
#pragma unroll
    for (int t = 0; t < CT; ++t)
#pragma unroll
      for (int r = 0; r < 8; ++r) acc[t][r] = 0.f;
  }

  // Prefetch the entire per-lane A strip into registers (one load clause).
  const float* __restrict__ xrow = X + (size_t)arow * K;
  v2f afr[K4];
#pragma unroll
  for (int i = 0; i < K4; ++i) {
    const int ka = i * 4 + half * 2;
    afr[i].x = xrow[ka];
    afr[i].y = xrow[ka + 1];
  }

  // Fully-unrolled WMMA chain: B fragments from LDS (ds_load_b64).
#pragma unroll
  for (int i = 0; i < K4; ++i) {
    const int kp = i * 2 + half;
#pragma unroll
    for (int t = 0; t < CT; ++t) {
      const float2 bf = wlds[kp * Mpad + bcol[t]];
      v2f b;
      b.x = bf.x;
      b.y = bf.y;
      acc[t] = __builtin_amdgcn_wmma_f32_16x16x4_f32(false, afr[i], false, b,
                                                     (short)0, acc[t], false, false);
    }
  }

#pragma unroll
  for (int t = 0; t < CT; ++t) {
    const int c = bcol[t];
    float add = 0.f;
    if (c < M) {
      if (bias1) add += bias1[c];
      if (bias2) add += bias2[c];
    }
#pragma unroll
    for (int r = 0; r < 8; ++r) {
      const int row = tm * 16 + half * 8 + r;
      if (row < nrows && c < M) Y[(size_t)row * M + c] = acc[t][r] + add;
    }
  }
}

// ---------------------------------------------------------------------
// Per-node attention scores: ss[i] = h[i]·a_src, sd[i] = h[i]·a_dst
// One wave per node, wave32 shuffle reduction.
// ---------------------------------------------------------------------
__global__ void node_scores(const float* __restrict__ h, const float* __restrict__ asrc,
                            const float* __restrict__ adst, float* __restrict__ ss,
                            float* __restrict__ sd, int n, int Hdim) {
  const int lane = threadIdx.x & 31;
  const int node = (int)((blockIdx.x * (unsigned)blockDim.x + threadIdx.x) >> 5);
  if (node >= n) return;
  float a = 0.f, b = 0.f;
  for (int c = lane; c < Hdim; c += 32) {
    const float v = h[(size_t)node * Hdim + c];
    a += v * asrc[c];
    b += v * adst[c];
  }
#pragma unroll
  for (int off = 16; off > 0; off >>= 1) {
    a += __shfl_xor(a, off, 32);
    b += __shfl_xor(b, off, 32);
  }
  if (lane == 0) { ss[node] = a; sd[node] = b; }
}

__global__ void fill_f32(float* __restrict__ p, float v, long n) {
  const long i = (long)blockIdx.x * blockDim.x + threadIdx.x;
  if (i < n) p[i] = v;
}

__global__ void bias_bcast(float* __restrict__ out, const float* __restrict__ b,
                           int n, int Hdim) {
  const long i = (long)blockIdx.x * blockDim.x + threadIdx.x;
  if (i < (long)n * Hdim) out[i] = b[i % Hdim];
}

// edge j in [0, E+n): j < E -> (src,dst) from edge_index; else self-loop (j-E, j-E)
__global__ void gat_edge_logits(const int* __restrict__ ei, const float* __restrict__ ss,
                                const float* __restrict__ sd, float* __restrict__ e,
                                float* __restrict__ emax, int E_, int n) {
  const int j = blockIdx.x * blockDim.x + threadIdx.x;
  if (j >= E_ + n) return;
  const int s = (j < E_) ? ei[j] : (j - E_);
  const int d = (j < E_) ? ei[E_ + j] : (j - E_);
  float v = ss[s] + sd[d];
  v = (v > 0.f) ? v : NEG_SLOPE * v;
  e[j] = v;
  atomicMaxF(&emax[d], v);
}

__global__ void gat_edge_softmax_num(const int* __restrict__ ei, float* __restrict__ e,
                                     const float* __restrict__ emax, float* __restrict__ denom,
                                     int E_, int n) {
  const int j = blockIdx.x * blockDim.x + threadIdx.x;
  if (j >= E_ + n) return;
  const int d = (j < E_) ? ei[E_ + j] : (j - E_);
  const float p = __expf(e[j] - emax[d]);
  e[j] = p;
  atomicAdd(&denom[d], p);
}

// wave per edge: out[dst] += alpha * h[src]  (lane-coalesced atomics)
__global__ void gat_edge_scatter(const int* __restrict__ ei, const float* __restrict__ e,
                                 const float* __restrict__ denom, const float* __restrict__ h,
                                 float* __restrict__ out, int E_, int n, int Hdim) {
  const int lane = threadIdx.x & 31;
  const int j = (int)((blockIdx.x * (unsigned)blockDim.x + threadIdx.x) >> 5);
  if (j >= E_ + n) return;
  const int s = (j < E_) ? ei[j] : (j - E_);
  const int d = (j < E_) ? ei[E_ + j] : (j - E_);
  const float alpha = e[j] / denom[d];
  for (int c = lane; c < Hdim; c += 32)
    atomicAdd(&out[(size_t)d * Hdim + c], alpha * h[(size_t)s * Hdim + c]);
}

__global__ void deg_count(const int* __restrict__ ei, float* __restrict__ deg,
                          int E_, int n, int include_self) {
  const int j = blockIdx.x * blockDim.x + threadIdx.x;
  const int tot = E_ + (include_self ? n : 0);
  if (j >= tot) return;
  const int d = (j < E_) ? ei[E_ + j] : (j - E_);
  atomicAdd(&deg[d], 1.0f);
}

// wave per edge (incl self loops): out[dst] += rsqrt(deg[s])*rsqrt(deg[d]) * h[src]
__global__ void gcn_edge_scatter(const int* __restrict__ ei, const float* __restrict__ deg,
                                 const float* __restrict__ h, float* __restrict__ out,
                                 int E_, int n, int Hdim) {
  const int lane = threadIdx.x & 31;
  const int j = (int)((blockIdx.x * (unsigned)blockDim.x + threadIdx.x) >> 5);
  if (j >= E_ + n) return;
  const int s = (j < E_) ? ei[j] : (j - E_);
  const int d = (j < E_) ? ei[E_ + j] : (j - E_);
  const float w = rsqrtf(deg[s]) * rsqrtf(deg[d]);
  for (int c = lane; c < Hdim; c += 32)
    atomicAdd(&out[(size_t)d * Hdim + c], w * h[(size_t)s * Hdim + c]);
}

// wave per edge (no self loops): out[dst] += h[src]
__global__ void sage_edge_scatter(const int* __restrict__ ei, const float* __restrict__ h,
                                  float* __restrict__ out, int E_, int Hdim) {
  const int lane = threadIdx.x & 31;
  const int j = (int)((blockIdx.x * (unsigned)blockDim.x + threadIdx.x) >> 5);
  if (j >= E_) return;
  const int s = ei[j];
  const int d = ei[E_ + j];
  for (int c = lane; c < Hdim; c += 32)
    atomicAdd(&out[(size_t)d * Hdim + c], h[(size_t)s * Hdim + c]);
}

__global__ void sage_divide(const float* __restrict__ cnt, float* __restrict__ agg,
                            int n, int Hdim) {
  const int lane = threadIdx.x & 31;
  const int node = (int)((blockIdx.x * (unsigned)blockDim.x + threadIdx.x) >> 5);
  if (node >= n) return;
  const float sc = 1.0f / fmaxf(cnt[node], 1.0f);
  for (int c = lane; c < Hdim; c += 32)
    agg[(size_t)node * Hdim + c] *= sc;
}

// Fused ELU + LayerNorm; one wave per node, Hdim <= 96.
__global__ void elu_ln(const float* __restrict__ x, const float* __restrict__ g,
                       const float* __restrict__ beta, float* __restrict__ y,
                       int n, int Hdim) {
  const int lane = threadIdx.x & 31;
  const int node = (int)((blockIdx.x * (unsigned)blockDim.x + threadIdx.x) >> 5);
  if (node >= n) return;
  float v[3];
  float sum = 0.f;
#pragma unroll
  for (int i = 0; i < 3; ++i) {
    const int c = lane + i * 32;
    float t = (c < Hdim) ? x[(size_t)node * Hdim + c] : 0.f;
    t = eluf(t);
    v[i] = t;
    if (c < Hdim) sum += t;
  }
#pragma unroll
  for (int off = 16; off > 0; off >>= 1) sum += __shfl_xor(sum, off, 32);
  const float mean = sum / (float)Hdim;
  float var = 0.f;
#pragma unroll
  for (int i = 0; i < 3; ++i) {
    const int c = lane + i * 32;
    if (c < Hdim) { const float dlt = v[i] - mean; var += dlt * dlt; }
  }
#pragma unroll
  for (int off = 16; off > 0; off >>= 1) var += __shfl_xor(var, off, 32);
  const float inv = rsqrtf(var / (float)Hdim + LN_EPS);
#pragma unroll
  for (int i = 0; i < 3; ++i) {
    const int c = lane + i * 32;
    if (c < Hdim)
      y[(size_t)node * Hdim + c] = (v[i] - mean) * inv * g[c] + beta[c];
  }
}

// =====================================================================
extern "C" void kernel_launch(void* const* d_in, const int* in_sizes, int n_in,
                              void* d_out, int out_size, void* d_ws, size_t ws_size,
                              hipStream_t stream) {
  (void)in_sizes; (void)n_in; (void)out_size; (void)ws_size;

  const float* x    = (const float*)d_in[0];
  const int*   ei   = (const int*)d_in[1];     // [2,E] row-major: src row then dst row
  const float* W1   = (const float*)d_in[2];
  const float* as1  = (const float*)d_in[3];
  const float* ad1  = (const float*)d_in[4];
  const float* b1   = (const float*)d_in[5];
  const float* g1   = (const float*)d_in[6];
  const float* be1  = (const float*)d_in[7];
  const float* Wg   = (const float*)d_in[8];
  const float* bg   = (const float*)d_in[9];
  const float* g2   = (const float*)d_in[10];
  const float* be2  = (const float*)d_in[11];
  const float* Wl   = (const float*)d_in[12];
  const float* Wr   = (const float*)d_in[13];
  const float* bs   = (const float*)d_in[14];
  const float* g3   = (const float*)d_in[15];
  const float* be3  = (const float*)d_in[16];
  const float* W4   = (const float*)d_in[17];
  const float* as4  = (const float*)d_in[18];
  const float* ad4  = (const float*)d_in[19];
  const float* b4   = (const float*)d_in[20];
  const float* Wres = (const float*)d_in[21];
  const float* bres = (const float*)d_in[22];

  float* inter = (float*)d_out;                 // output 0: [N,H]
  float* out2  = inter + (size_t)Nn * Hh;       // output 1: [N,C]

  // Workspace layout (floats); requires ~62 MB of d_ws.
  float* ws   = (float*)d_ws;
  float* buf0 = ws;                             // N*H
  float* buf1 = buf0 + (size_t)Nn * Hh;         // N*H
  float* buf2 = buf1 + (size_t)Nn * Hh;         // N*H
  float* ssb  = buf2 + (size_t)Nn * Hh;         // N
  float* sdb  = ssb + Nn;                       // N
  float* emax = sdb + Nn;                       // N
  float* den  = emax + Nn;                      // N (also deg / cnt)
  float* ebuf = den + Nn;                       // E+N (logits then softmax numerators)
  float* h4   = buf0;                           // alias: buf0 is free in stage 4

  auto cdiv = [](long a, long b) { return (int)((a + b - 1) / b); };
  const int  BT = 256;
  const int  EN = Ee + Nn;
  const float NEG_INF = -__builtin_inff();

  const int rowTiles   = (Nn + 15) / 16;
  const int gemmBlocks = cdiv(rowTiles, BT / 32);

  // CT = ceil(M/16); K in {128, 96} as template params.
  auto gemm = [&](const float* Xp, const float* Wp, float* Yp, const float* bb1,
                  const float* bb2, int K, int M, int acc) {
    if (M == Hh) {
      if (K == Ff)
        gemm_wmma_f32<6, 128><<<gemmBlocks, BT, 0, stream>>>(Xp, Wp, Yp, bb1, bb2, Nn, M, acc);
      else
        gemm_wmma_f32<6, 96><<<gemmBlocks, BT, 0, stream>>>(Xp, Wp, Yp, bb1, bb2, Nn, M, acc);
    } else {
      if (K == Ff)
        gemm_wmma_f32<3, 128><<<gemmBlocks, BT, 0, stream>>>(Xp, Wp, Yp, bb1, bb2, Nn, M, acc);
      else
        gemm_wmma_f32<3, 96><<<gemmBlocks, BT, 0, stream>>>(Xp, Wp, Yp, bb1, bb2, Nn, M, acc);
    }
  };

  // ---------------- Layer 1: GAT(F->H), ELU, LN ----------------
  gemm(x, W1, buf0, nullptr, nullptr, Ff, Hh, 0);                           // h1 = x@W1
  node_scores<<<cdiv((long)Nn * 32, BT), BT, 0, stream>>>(buf0, as1, ad1, ssb, sdb, Nn, Hh);
  fill_f32<<<cdiv(Nn, BT), BT, 0, stream>>>(emax, NEG_INF, Nn);
  fill_f32<<<cdiv(Nn, BT), BT, 0, stream>>>(den, 0.f, Nn);
  bias_bcast<<<cdiv((long)Nn * Hh, BT), BT, 0, stream>>>(buf1, b1, Nn, Hh); // out1 = b1
  gat_edge_logits<<<cdiv(EN, BT), BT, 0, stream>>>(ei, ssb, sdb, ebuf, emax, Ee, Nn);
  gat_edge_softmax_num<<<cdiv(EN, BT), BT, 0, stream>>>(ei, ebuf, emax, den, Ee, Nn);
  gat_edge_scatter<<<cdiv((long)EN * 32, BT), BT, 0, stream>>>(ei, ebuf, den, buf0, buf1, Ee, Nn, Hh);
  elu_ln<<<cdiv((long)Nn * 32, BT), BT, 0, stream>>>(buf1, g1, be1, buf2, Nn, Hh);

  // ---------------- Layer 2: GCN(H->H), ELU, LN ----------------
  gemm(buf2, Wg, buf0, nullptr, nullptr, Hh, Hh, 0);                        // h2 = ln1@Wg
  fill_f32<<<cdiv(Nn, BT), BT, 0, stream>>>(den, 0.f, Nn);
  deg_count<<<cdiv(EN, BT), BT, 0, stream>>>(ei, den, Ee, Nn, 1);
  bias_bcast<<<cdiv((long)Nn * Hh, BT), BT, 0, stream>>>(buf1, bg, Nn, Hh);
  gcn_edge_scatter<<<cdiv((long)EN * 32, BT), BT, 0, stream>>>(ei, den, buf0, buf1, Ee, Nn, Hh);
  elu_ln<<<cdiv((long)Nn * 32, BT), BT, 0, stream>>>(buf1, g2, be2, buf2, Nn, Hh);

  // ---------------- Layer 3: SAGE(H->H), ELU, LN -> intermediate ----------------
  fill_f32<<<cdiv(Nn, BT), BT, 0, stream>>>(den, 0.f, Nn);
  deg_count<<<cdiv(Ee, BT), BT, 0, stream>>>(ei, den, Ee, Nn, 0);           // no self loops
  fill_f32<<<cdiv((long)Nn * Hh, BT), BT, 0, stream>>>(buf1, 0.f, (long)Nn * Hh);
  sage_edge_scatter<<<cdiv((long)Ee * 32, BT), BT, 0, stream>>>(ei, buf2, buf1, Ee, Hh);
  sage_divide<<<cdiv((long)Nn * 32, BT), BT, 0, stream>>>(den, buf1, Nn, Hh);
  gemm(buf1, Wl, buf0, bs, nullptr, Hh, Hh, 0);                             // agg@Wl + bs
  gemm(buf2, Wr, buf0, nullptr, nullptr, Hh, Hh, 1);                        // += ln2@Wr
  elu_ln<<<cdiv((long)Nn * 32, BT), BT, 0, stream>>>(buf0, g3, be3, inter, Nn, Hh);

  // ---------------- Layer 4: GAT(H->C) + residual ----------------
  gemm(inter, W4, h4, nullptr, nullptr, Hh, Cc, 0);                         // h4 = inter@W4
  node_scores<<<cdiv((long)Nn * 32, BT), BT, 0, stream>>>(h4, as4, ad4, ssb, sdb, Nn, Cc);
  fill_f32<<<cdiv(Nn, BT), BT, 0, stream>>>(emax, NEG_INF, Nn);
  fill_f32<<<cdiv(Nn, BT), BT, 0, stream>>>(den, 0.f, Nn);
  gat_edge_logits<<<cdiv(EN, BT), BT, 0, stream>>>(ei, ssb, sdb, ebuf, emax, Ee, Nn);
  gat_edge_softmax_num<<<cdiv(EN, BT), BT, 0, stream>>>(ei, ebuf, emax, den, Ee, Nn);
  gemm(x, Wres, out2, bres, b4, Ff, Cc, 0);                                 // out2 = x@Wres + bres + b4
  gat_edge_scatter<<<cdiv((long)EN * 32, BT), BT, 0, stream>>>(ei, ebuf, den, h4, out2, Ee, Nn, Cc);
}